// MyColorHistogram_2327872275016
// MI455X (gfx1250) — compile-verified
//
#include <hip/hip_runtime.h>

// ---------------------------------------------------------------------------
// Per-(image,channel) 64-bin histogram, L1-normalized.
// B=64, C=3, H=512, W=512, bins=64  ->  192 planes of 262144 pixels each.
// HBM-bound: 201 MB read once => ~8.6 us floor at 23.3 TB/s.
// Strategy: b128 vector loads + wave32-private LDS histograms (ds_add_u32),
// per-WG merge via global_atomic_add_u32, then a deterministic scale by
// 1/(H*W) (sum of every histogram is exactly H*W = 2^18; +1e-7 is below ulp).
// ---------------------------------------------------------------------------

#define BINS           64
#define WG_THREADS     256            // 8 wave32 per workgroup
#define WAVES_PER_WG   (WG_THREADS / 32)
#define VEC_PER_THREAD 16             // float4 per thread
#define LOAD_BATCH     8              // independent b128 loads in flight
// pixels per WG = 256 * 16 * 4 = 16384 ; plane = 262144 px -> 16 chunks/plane

__global__ void hist_zero(unsigned int* __restrict__ ws, int n) {
    int i = blockIdx.x * blockDim.x + threadIdx.x;
    if (i < n) ws[i] = 0u;
}

__device__ __forceinline__ int bin_of(float x) {
    // clamp(x,0,1); floor(x*64); clamp to [0,63].
    // min(x,1)*64 then cvt (trunc) then med3(i,0,63) handles both x<0 and x==1.
    float t = __builtin_fminf(x, 1.0f) * 64.0f;
    int   i = (int)t;                               // v_cvt_i32_f32 (trunc)
    i = i < 0 ? 0 : i;                              // med3 with next line
    return i > (BINS - 1) ? (BINS - 1) : i;         // -> v_med3_i32 i,0,63
}

__global__ __launch_bounds__(WG_THREADS)
void hist_kernel(const float* __restrict__ x,
                 unsigned int* __restrict__ counts,
                 long long pixels_per_plane) {
    __shared__ unsigned int lh[WAVES_PER_WG * BINS];

    const int tid = threadIdx.x;

    // zero LDS sub-histograms
    #pragma unroll
    for (int i = tid; i < WAVES_PER_WG * BINS; i += WG_THREADS) lh[i] = 0u;
    __syncthreads();

    const int       plane = blockIdx.y;   // 0..191 : (b*C + c)
    const int       chunk = blockIdx.x;   // 0..15  : tile within plane
    const long long plane_base = (long long)plane * pixels_per_plane;

    const float4* __restrict__ p =
        (const float4*)(x + plane_base) +
        (long long)chunk * (WG_THREADS * VEC_PER_THREAD);

    unsigned int* __restrict__ myh = lh + (tid >> 5) * BINS;  // wave-private

    #pragma unroll
    for (int base = 0; base < VEC_PER_THREAD; base += LOAD_BATCH) {
        float4 d[LOAD_BATCH];
        // batch of independent coalesced global_load_b128
        #pragma unroll
        for (int v = 0; v < LOAD_BATCH; ++v)
            d[v] = p[(long long)(base + v) * WG_THREADS + tid];
        // scatter into wave-private LDS histogram (ds_add_u32)
        #pragma unroll
        for (int v = 0; v < LOAD_BATCH; ++v) {
            atomicAdd(&myh[bin_of(d[v].x)], 1u);
            atomicAdd(&myh[bin_of(d[v].y)], 1u);
            atomicAdd(&myh[bin_of(d[v].z)], 1u);
            atomicAdd(&myh[bin_of(d[v].w)], 1u);
        }
    }
    __syncthreads();

    // merge the 8 wave-private copies, one global atomic per bin per WG
    if (tid < BINS) {
        unsigned int s = 0;
        #pragma unroll
        for (int w = 0; w < WAVES_PER_WG; ++w) s += lh[w * BINS + tid];
        atomicAdd(&counts[plane * BINS + tid], s);   // global_atomic_add_u32
    }
}

__global__ void hist_norm(const unsigned int* __restrict__ counts,
                          float* __restrict__ out, int n, float inv) {
    int i = blockIdx.x * blockDim.x + threadIdx.x;
    if (i < n) out[i] = (float)counts[i] * inv;
}

extern "C" void kernel_launch(void* const* d_in, const int* in_sizes, int n_in,
                              void* d_out, int out_size, void* d_ws, size_t ws_size,
                              hipStream_t stream) {
    const float*  x      = (const float*)d_in[0];
    float*        out    = (float*)d_out;
    unsigned int* counts = (unsigned int*)d_ws;

    const int nhist  = out_size;            // B * C * bins = 12288
    const int planes = nhist / BINS;        // 192
    const long long total     = (long long)in_sizes[0];      // 50,331,648
    const long long per_plane = total / planes;              // 262,144
    const int chunks =
        (int)(per_plane / (4LL * WG_THREADS * VEC_PER_THREAD)); // 16

    // 1) zero counters (d_ws is poisoned; must re-zero every launch)
    hist_zero<<<(nhist + 255) / 256, 256, 0, stream>>>(counts, nhist);

    // 2) histogram: 16 x 192 = 3072 workgroups, 8 waves each
    dim3 grid(chunks, planes);
    hist_kernel<<<grid, WG_THREADS, 0, stream>>>(x, counts, per_plane);

    // 3) normalize: sum per histogram is exactly per_plane; +1e-7 < ulp(2^18)
    const float inv = 1.0f / ((float)per_plane + 1e-7f);     // == 2^-18 exactly
    hist_norm<<<(nhist + 255) / 256, 256, 0, stream>>>(counts, out, nhist, inv);
}